// VisGNN_1915555414183
// MI455X (gfx1250) — compile-verified
//
#include <hip/hip_runtime.h>
#include <hip/hip_bf16.h>
#include <math.h>

// ---------------------------------------------------------------------------
// VisGNN on gfx1250 (MI455X, wave32).
//
// Factorized CGConv: z@W = x_dst@W_top + x_src@W_mid + e*W_last, so all dense
// work becomes node-level GEMMs (N x 128 @ 128 x 128) done with
// V_WMMA_F32_16X16X4_F32 (fp32-exact). Edge phase is gather + elementwise
// gates + atomic scatter; the four gathered tables (102 MB) sit in the 192 MB
// L2, so random access stays on-die.
//
// GEMM tiling: one wave owns a 16x64 output tile (4 WMMA accumulators).
// Per k-step: one A-fragment load reused across 4 column tiles, and 4
// independent WMMA chains to hide matrix-pipe latency.
// ---------------------------------------------------------------------------

typedef float v2f __attribute__((ext_vector_type(2)));
typedef float v8f __attribute__((ext_vector_type(8)));

#define HDIM 128

// ---------------------------------------------------------------------------
// fp32 WMMA GEMM:  Y[n x 128] = X[n x K] @ W[K x 128] (+bias) (+relu)
// One wave per 16x64 tile; grid.x = 2 * ceil(n/16). K % 4 == 0.
//
// VGPR layouts per CDNA5 ISA 7.12.2:
//   A 16x4 f32 : lanes 0-15 -> {K=0,K=1}, lanes 16-31 -> {K=2,K=3}, M = lane&15
//   B 4x16 f32 : VGPR v, lanes 0-15 -> K=v, lanes 16-31 -> K=v+2, N = lane&15
//   C/D 16x16  : VGPR v, lanes 0-15 -> M=v, lanes 16-31 -> M=v+8, N = lane&15
// ---------------------------------------------------------------------------
__global__ void gemm_wmma_f32(const float* __restrict__ X, int ldx,
                              const float* __restrict__ W,    // K x 128, row-major
                              const float* __restrict__ bias, // may be null
                              float* __restrict__ Y,          // n x 128
                              int nrows, int K, int relu)
{
    const int lane = threadIdx.x & 31;
    const int tile = blockIdx.x;
    const int cg   = tile & 1;        // column group: 0 -> cols 0..63, 1 -> 64..127
    const int rt   = tile >> 1;
    const int row0 = rt * 16;
    const int col0 = cg * 64;
    const int half = lane >> 4;       // 0: K lanes {0,1}, 1: K lanes {2,3}
    const int l16  = lane & 15;
    const int arow = row0 + l16;      // A-fragment row held by this lane
    const int koff = half * 2;
    const bool arow_ok = (arow < nrows);

    const float* __restrict__ xrow = X + (size_t)arow * ldx;
    const float* __restrict__ wcol = W + col0 + l16;  // + t*16 per column tile

    v8f acc0 = {};
    v8f acc1 = {};
    v8f acc2 = {};
    v8f acc3 = {};

    for (int k = 0; k < K; k += 4) {
        v2f a;
        if (arow_ok) {
            const float2 xa = *(const float2*)(xrow + k + koff);
            a.x = xa.x; a.y = xa.y;
        } else {
            a = (v2f){0.f, 0.f};
        }
        const float* w0 = wcol + (size_t)(k + koff) * HDIM;
        const float* w1 = wcol + (size_t)(k + koff + 1) * HDIM;

        v2f b0, b1, b2, b3;
        b0.x = w0[0];  b0.y = w1[0];
        b1.x = w0[16]; b1.y = w1[16];
        b2.x = w0[32]; b2.y = w1[32];
        b3.x = w0[48]; b3.y = w1[48];

        acc0 = __builtin_amdgcn_wmma_f32_16x16x4_f32(
            false, a, false, b0, (short)0, acc0, false, false);
        acc1 = __builtin_amdgcn_wmma_f32_16x16x4_f32(
            false, a, false, b1, (short)0, acc1, false, false);
        acc2 = __builtin_amdgcn_wmma_f32_16x16x4_f32(
            false, a, false, b2, (short)0, acc2, false, false);
        acc3 = __builtin_amdgcn_wmma_f32_16x16x4_f32(
            false, a, false, b3, (short)0, acc3, false, false);
    }

    float bv0 = 0.f, bv1 = 0.f, bv2 = 0.f, bv3 = 0.f;
    if (bias) {
        bv0 = bias[col0 + l16 +  0];
        bv1 = bias[col0 + l16 + 16];
        bv2 = bias[col0 + l16 + 32];
        bv3 = bias[col0 + l16 + 48];
    }
#pragma unroll
    for (int v = 0; v < 8; ++v) {
        const int r = row0 + v + half * 8;
        if (r < nrows) {
            float* yr = Y + (size_t)r * HDIM + col0 + l16;
            float v0 = acc0[v] + bv0;
            float v1 = acc1[v] + bv1;
            float v2 = acc2[v] + bv2;
            float v3 = acc3[v] + bv3;
            if (relu) {
                v0 = fmaxf(v0, 0.f); v1 = fmaxf(v1, 0.f);
                v2 = fmaxf(v2, 0.f); v3 = fmaxf(v3, 0.f);
            }
            yr[0]  = v0;
            yr[16] = v1;
            yr[32] = v2;
            yr[48] = v3;
        }
    }
}

// ---------------------------------------------------------------------------
// Utility / edge-phase kernels
// ---------------------------------------------------------------------------
__global__ void fill_zero_k(float* __restrict__ p, size_t n)
{
    size_t i = (size_t)blockIdx.x * blockDim.x + threadIdx.x;
    if (i < n) p[i] = 0.f;
}

__global__ void count_edges_k(const int* __restrict__ dst,
                              float* __restrict__ cnt, int E)
{
    int e = blockIdx.x * blockDim.x + threadIdx.x;
    if (e < E) atomicAdd(&cnt[dst[e]], 1.0f);
}

__global__ void finalize_deg_k(const float* __restrict__ cnt,
                               float* __restrict__ dinv,
                               float* __restrict__ dege, int N)
{
    int i = blockIdx.x * blockDim.x + threadIdx.x;
    if (i < N) {
        dinv[i] = rsqrtf(cnt[i] + 1.0f);   // GCN: deg includes self-loop
        dege[i] = fmaxf(cnt[i], 1.0f);     // CGConv mean denominator
    }
}

// GCN aggregation: G[dst] += h[src] * dinv[src]*dinv[dst]
// 32 lanes per edge, 4 floats per lane.
__global__ void gcn_agg_k(const float* __restrict__ h,
                          const int* __restrict__ src,
                          const int* __restrict__ dst,
                          const float* __restrict__ dinv,
                          float* __restrict__ G, int E)
{
    int gid = blockIdx.x * blockDim.x + threadIdx.x;
    int e = gid >> 5;
    if (e >= E) return;
    int l = (gid & 31) * 4;
    int s = src[e], d = dst[e];
    float nrm = dinv[s] * dinv[d];
    const float4 hv = *(const float4*)(h + (size_t)s * HDIM + l);
    float* gp = G + (size_t)d * HDIM + l;
    atomicAdd(gp + 0, hv.x * nrm);
    atomicAdd(gp + 1, hv.y * nrm);
    atomicAdd(gp + 2, hv.z * nrm);
    atomicAdd(gp + 3, hv.w * nrm);
}

// H = relu(G + h0 * dinv^2 + b1)
__global__ void gcn1_combine_k(const float* __restrict__ G,
                               const float* __restrict__ h0,
                               const float* __restrict__ dinv,
                               const float* __restrict__ b1,
                               float* __restrict__ H, int N)
{
    size_t i = (size_t)blockIdx.x * blockDim.x + threadIdx.x;
    if (i >= (size_t)N * HDIM) return;
    int node = (int)(i >> 7);
    int c = (int)(i & (HDIM - 1));
    float s = dinv[node];
    H[i] = fmaxf(G[i] + h0[i] * (s * s) + b1[c], 0.f);
}

__device__ __forceinline__ float sigmoid_f(float x)
{
    return 1.0f / (1.0f + expf(-x));
}

__device__ __forceinline__ float softplus_f(float x)
{
    return (x > 20.0f) ? x : log1pf(expf(x));
}

// CGConv edge phase: G[dst] += sigmoid(P[d]+Q[s]+a*wfE) * softplus(R[d]+S[s]+a*wsE)
__global__ void cg_edge_k(const float* __restrict__ P,
                          const float* __restrict__ Q,
                          const float* __restrict__ R,
                          const float* __restrict__ S,
                          const float* __restrict__ wfE,
                          const float* __restrict__ wsE,
                          const int* __restrict__ src,
                          const int* __restrict__ dst,
                          const float* __restrict__ ea,
                          float* __restrict__ G, int E)
{
    int gid = blockIdx.x * blockDim.x + threadIdx.x;
    int e = gid >> 5;
    if (e >= E) return;
    int l = (gid & 31) * 4;
    int s = src[e], d = dst[e];
    float a = ea[e];

    const float4 p  = *(const float4*)(P + (size_t)d * HDIM + l);
    const float4 q  = *(const float4*)(Q + (size_t)s * HDIM + l);
    const float4 r  = *(const float4*)(R + (size_t)d * HDIM + l);
    const float4 ss = *(const float4*)(S + (size_t)s * HDIM + l);
    const float4 wf = *(const float4*)(wfE + l);
    const float4 ws = *(const float4*)(wsE + l);

    float m0 = sigmoid_f(p.x + q.x + a * wf.x) * softplus_f(r.x + ss.x + a * ws.x);
    float m1 = sigmoid_f(p.y + q.y + a * wf.y) * softplus_f(r.y + ss.y + a * ws.y);
    float m2 = sigmoid_f(p.z + q.z + a * wf.z) * softplus_f(r.z + ss.z + a * ws.z);
    float m3 = sigmoid_f(p.w + q.w + a * wf.w) * softplus_f(r.w + ss.w + a * ws.w);

    float* gp = G + (size_t)d * HDIM + l;
    atomicAdd(gp + 0, m0);
    atomicAdd(gp + 1, m1);
    atomicAdd(gp + 2, m2);
    atomicAdd(gp + 3, m3);
}

// H = relu(H + G / dege)   (in-place, elementwise)
__global__ void cg_combine_k(float* __restrict__ H,
                             const float* __restrict__ G,
                             const float* __restrict__ dege, int N)
{
    size_t i = (size_t)blockIdx.x * blockDim.x + threadIdx.x;
    if (i >= (size_t)N * HDIM) return;
    int node = (int)(i >> 7);
    H[i] = fmaxf(H[i] + G[i] / dege[node], 0.f);
}

// hw2[n][0..1] = H[n] @ W2 (128 x 2). One wave per node, shuffle reduction.
__global__ void node_w2_k(const float* __restrict__ H,
                          const float* __restrict__ W2,
                          float* __restrict__ hw2, int N)
{
    int node = blockIdx.x * (blockDim.x >> 5) + (threadIdx.x >> 5);
    int l = threadIdx.x & 31;
    if (node >= N) return;

    const float4 h   = *(const float4*)(H + (size_t)node * HDIM + l * 4);
    const float4 w01 = *(const float4*)(W2 + (size_t)l * 8);      // c0k0 c0k1 c1k0 c1k1
    const float4 w23 = *(const float4*)(W2 + (size_t)l * 8 + 4);  // c2k0 c2k1 c3k0 c3k1

    float a0 = h.x * w01.x + h.y * w01.z + h.z * w23.x + h.w * w23.z;
    float a1 = h.x * w01.y + h.y * w01.w + h.z * w23.y + h.w * w23.w;

#pragma unroll
    for (int off = 16; off > 0; off >>= 1) {
        a0 += __shfl_xor(a0, off, 32);
        a1 += __shfl_xor(a1, off, 32);
    }
    if (l == 0) {
        hw2[(size_t)node * 2 + 0] = a0;
        hw2[(size_t)node * 2 + 1] = a1;
    }
}

// out[dst] += hw2[src] * norm   (D_OUT = 2)
__global__ void gcn2_edge_k(const float* __restrict__ hw2,
                            const int* __restrict__ src,
                            const int* __restrict__ dst,
                            const float* __restrict__ dinv,
                            float* __restrict__ out, int E)
{
    int e = blockIdx.x * blockDim.x + threadIdx.x;
    if (e >= E) return;
    int s = src[e], d = dst[e];
    float nrm = dinv[s] * dinv[d];
    atomicAdd(&out[(size_t)d * 2 + 0], hw2[(size_t)s * 2 + 0] * nrm);
    atomicAdd(&out[(size_t)d * 2 + 1], hw2[(size_t)s * 2 + 1] * nrm);
}

// out += hw2 * dinv^2 + b2
__global__ void gcn2_final_k(float* __restrict__ out,
                             const float* __restrict__ hw2,
                             const float* __restrict__ dinv,
                             const float* __restrict__ b2, int N)
{
    int i = blockIdx.x * blockDim.x + threadIdx.x;
    if (i >= N * 2) return;
    int node = i >> 1;
    int k = i & 1;
    float s = dinv[node];
    out[i] = out[i] + hw2[i] * (s * s) + b2[k];
}

// ---------------------------------------------------------------------------
// Host-side orchestration (graph-capture safe: only stream launches)
// ---------------------------------------------------------------------------
extern "C" void kernel_launch(void* const* d_in, const int* in_sizes, int n_in,
                              void* d_out, int out_size, void* d_ws, size_t ws_size,
                              hipStream_t stream)
{
    const float* x   = (const float*)d_in[0];
    const int*   ei  = (const int*)  d_in[1];
    const float* ea  = (const float*)d_in[2];
    const float* W1  = (const float*)d_in[3];
    const float* b1  = (const float*)d_in[4];
    const float* Wf0 = (const float*)d_in[5];
    const float* bf0 = (const float*)d_in[6];
    const float* Ws0 = (const float*)d_in[7];
    const float* bs0 = (const float*)d_in[8];
    const float* Wf1 = (const float*)d_in[9];
    const float* bf1 = (const float*)d_in[10];
    const float* Ws1 = (const float*)d_in[11];
    const float* bs1 = (const float*)d_in[12];
    const float* W2  = (const float*)d_in[13];
    const float* b2  = (const float*)d_in[14];
    float* out = (float*)d_out;

    const int N = in_sizes[0] / 64;   // x is N x 64
    const int E = in_sizes[1] / 2;    // edge_index is 2 x E
    const int* src = ei;
    const int* dst = ei + E;

    // workspace layout (floats): H, G, P, Q, R, S  (each N*128), then vectors
    float* ws = (float*)d_ws;
    const size_t NH = (size_t)N * HDIM;
    float* H    = ws;
    float* G    = ws + 1 * NH;
    float* P    = ws + 2 * NH;
    float* Q    = ws + 3 * NH;
    float* R    = ws + 4 * NH;
    float* S    = ws + 5 * NH;
    float* cnt  = ws + 6 * NH;
    float* dinv = cnt + N;
    float* dege = dinv + N;
    float* hw2  = dege + N;           // N*2 floats

    const int TB = 256;
    auto nb = [](size_t n, int tb) { return (unsigned)((n + (size_t)tb - 1) / (size_t)tb); };

    const int row_tiles = (N + 15) / 16;
    const dim3 ggrid((unsigned)(2 * row_tiles));   // 2 column groups of 64
    const dim3 gblk(32);

    // --- degrees ---
    fill_zero_k<<<nb(N, TB), TB, 0, stream>>>(cnt, (size_t)N);
    fill_zero_k<<<nb((size_t)N * 2, TB), TB, 0, stream>>>(out, (size_t)N * 2);
    count_edges_k<<<nb(E, TB), TB, 0, stream>>>(dst, cnt, E);
    finalize_deg_k<<<nb(N, TB), TB, 0, stream>>>(cnt, dinv, dege, N);

    // --- GCN layer 1: h0 = x @ W1 (into P), aggregate, combine+relu -> H ---
    gemm_wmma_f32<<<ggrid, gblk, 0, stream>>>(x, 64, W1, nullptr, P, N, 64, 0);
    fill_zero_k<<<nb(NH, TB), TB, 0, stream>>>(G, NH);
    gcn_agg_k<<<nb((size_t)E * 32, TB), TB, 0, stream>>>(P, src, dst, dinv, G, E);
    gcn1_combine_k<<<nb(NH, TB), TB, 0, stream>>>(G, P, dinv, b1, H, N);

    // --- CGConv layers (factorized) ---
    const float* Wf[2] = {Wf0, Wf1};
    const float* Ws[2] = {Ws0, Ws1};
    const float* bf[2] = {bf0, bf1};
    const float* bs[2] = {bs0, bs1};
    for (int layer = 0; layer < 2; ++layer) {
        const float* wf  = Wf[layer];
        const float* wsm = Ws[layer];
        gemm_wmma_f32<<<ggrid, gblk, 0, stream>>>(H, HDIM, wf,               bf[layer], P, N, HDIM, 0);
        gemm_wmma_f32<<<ggrid, gblk, 0, stream>>>(H, HDIM, wf  + 128 * HDIM, nullptr,   Q, N, HDIM, 0);
        gemm_wmma_f32<<<ggrid, gblk, 0, stream>>>(H, HDIM, wsm,              bs[layer], R, N, HDIM, 0);
        gemm_wmma_f32<<<ggrid, gblk, 0, stream>>>(H, HDIM, wsm + 128 * HDIM, nullptr,   S, N, HDIM, 0);
        fill_zero_k<<<nb(NH, TB), TB, 0, stream>>>(G, NH);
        cg_edge_k<<<nb((size_t)E * 32, TB), TB, 0, stream>>>(
            P, Q, R, S, wf + 256 * HDIM, wsm + 256 * HDIM, src, dst, ea, G, E);
        cg_combine_k<<<nb(NH, TB), TB, 0, stream>>>(H, G, dege, N);
    }

    // --- GCN layer 2 (128 -> 2) ---
    node_w2_k<<<nb((size_t)N * 32, TB), TB, 0, stream>>>(H, W2, hw2, N);
    gcn2_edge_k<<<nb(E, TB), TB, 0, stream>>>(hw2, src, dst, dinv, out, E);
    gcn2_final_k<<<nb((size_t)N * 2, TB), TB, 0, stream>>>(out, hw2, dinv, b2, N);
}